// Geodesic_GNN_25890062860778
// MI455X (gfx1250) — compile-verified
//
#include <hip/hip_runtime.h>
#include <math.h>

// ---------------------------------------------------------------------------
// Geodesic GNN layer for MI455X (gfx1250, wave32).
//  K1: node projections (VALU, tiny)
//  K2: per-(edge,channel) message + global f32 atomics into L2-resident aggr
//      (edge streams read with NT hints so they don't evict the hot
//       node-projection / aggregation region from the 192MB L2)
//  K3: GVPerceptronVN via V_WMMA_F32_16X16X4_F32 (fp32 WMMA, K=16 in 4 steps)
// Workspace layout (floats): node_ss[N*16] | node_sv[N*16] | node_vv[N*48]
//                            | aggr_sca[N*16] | aggr_vec[N*48]   (= N*144 f32)
// ---------------------------------------------------------------------------

typedef __attribute__((ext_vector_type(2))) float v2f;
typedef __attribute__((ext_vector_type(8))) float v8f;

#define WAVES_PER_BLOCK 8

__device__ __forceinline__ float ldnt(const float* p) {
  return __builtin_nontemporal_load(p);
}
__device__ __forceinline__ void stnt(float* p, float v) {
  __builtin_nontemporal_store(v, p);
}

__device__ __forceinline__ v8f wmma4(v2f a, v2f b, v8f c) {
  // D(16x16,f32) = A(16x4,f32) * B(4x16,f32) + C
  return __builtin_amdgcn_wmma_f32_16x16x4_f32(
      /*neg_a=*/false, a, /*neg_b=*/false, b,
      /*c_mod=*/(short)0, c, /*reuse_a=*/false, /*reuse_b=*/false);
}

// 16x16x16 f32 matmul: 4 chained K=4 WMMAs. W is [16 x 16] row-major (k-major).
__device__ __forceinline__ v8f mm16(const v2f a[4], const float* __restrict__ W,
                                    int l16, int hi, v8f c) {
#pragma unroll
  for (int q = 0; q < 4; ++q) {
    const int k0 = 4 * q + 2 * hi;
    v2f b;
    b.x = W[k0 * 16 + l16];
    b.y = W[(k0 + 1) * 16 + l16];
    c = wmma4(a[q], b, c);
  }
  return c;
}

// ---------------------------------------------------------------------------
__global__ void zero_kernel(float* __restrict__ p, long n) {
  long i = (long)blockIdx.x * blockDim.x + threadIdx.x;
  const long stride = (long)gridDim.x * blockDim.x;
  for (; i < n; i += stride) p[i] = 0.0f;
}

// ---------------------------------------------------------------------------
// K1: one thread per (node, out-channel o)
__global__ __launch_bounds__(256) void node_proj_kernel(
    const float* __restrict__ node_sca, const float* __restrict__ node_vec,
    const float* __restrict__ W_nss, const float* __restrict__ b_nss,
    const float* __restrict__ W_nsv, const float* __restrict__ b_nsv,
    const float* __restrict__ W_nvv,
    float* __restrict__ node_ss, float* __restrict__ node_sv,
    float* __restrict__ node_vv, int N) {
  const int gid = blockIdx.x * blockDim.x + threadIdx.x;
  if (gid >= N * 16) return;
  const int n = gid >> 4;
  const int o = gid & 15;

  const float* xs = node_sca + (size_t)n * 4;
  float ss = b_nss[o], sv = b_nsv[o];
#pragma unroll
  for (int c = 0; c < 4; ++c) {
    const float x = xs[c];
    ss += x * W_nss[c * 16 + o];
    sv += x * W_nsv[c * 16 + o];
  }
  node_ss[gid] = ss;
  node_sv[gid] = sv;

  const float* xv = node_vec + (size_t)n * 9;  // [3 ch][3 comp]
  float v0 = 0.f, v1 = 0.f, v2 = 0.f;
#pragma unroll
  for (int c = 0; c < 3; ++c) {
    const float w = W_nvv[c * 16 + o];
    v0 += xv[c * 3 + 0] * w;
    v1 += xv[c * 3 + 1] * w;
    v2 += xv[c * 3 + 2] * w;
  }
  float* nv = node_vv + (size_t)n * 48 + o * 3;
  nv[0] = v0; nv[1] = v1; nv[2] = v2;
}

// ---------------------------------------------------------------------------
// K2: one thread per (edge, out-channel o); scatter with f32 atomics (L2).
// Edge streams (read exactly once) use non-temporal loads; the gathered
// node projections and the atomic targets stay regular-temporal (hot in L2).
__global__ __launch_bounds__(256) void edge_msg_kernel(
    const float* __restrict__ edge_sca, const float* __restrict__ edge_vec,
    const float* __restrict__ gds_dist, const int* __restrict__ edge_index,
    const float* __restrict__ W_ess, const float* __restrict__ b_ess,
    const float* __restrict__ W_esv, const float* __restrict__ b_esv,
    const float* __restrict__ W_evv,
    const float* __restrict__ node_ss, const float* __restrict__ node_sv,
    const float* __restrict__ node_vv,
    float* __restrict__ aggr_sca, float* __restrict__ aggr_vec, int E) {
  const int gid = blockIdx.x * blockDim.x + threadIdx.x;
  if (gid >= E * 16) return;
  const int e = gid >> 4;
  const int o = gid & 15;

  const float g = ldnt(gds_dist + e);
  float coeff = 0.0f;
  if (g >= 0.0f && g <= 10.0f)
    coeff = 0.5f * (cosf(g * 0.31415926535897932f) + 1.0f);
  if (coeff == 0.0f) return;  // exact-zero messages: skip atomics

  const int src = __builtin_nontemporal_load(edge_index + e);  // row 0 of [2,E]

  // edge scalar projections (6 -> 16)
  const float* es = edge_sca + (size_t)e * 6;
  float ess = b_ess[o], esv = b_esv[o];
#pragma unroll
  for (int c = 0; c < 6; ++c) {
    const float x = ldnt(es + c);
    ess += x * W_ess[c * 16 + o];
    esv += x * W_esv[c * 16 + o];
  }
  // edge vector projection (3 -> 16), per component
  const float* ev = edge_vec + (size_t)e * 9;
  float ev0 = 0.f, ev1 = 0.f, ev2 = 0.f;
#pragma unroll
  for (int c = 0; c < 3; ++c) {
    const float w = W_evv[c * 16 + o];
    ev0 += ldnt(ev + c * 3 + 0) * w;
    ev1 += ldnt(ev + c * 3 + 1) * w;
    ev2 += ldnt(ev + c * 3 + 2) * w;
  }

  const float nss = node_ss[(size_t)src * 16 + o];
  const float nsv = node_sv[(size_t)src * 16 + o];
  const float* nv = node_vv + (size_t)src * 48 + o * 3;

  atomicAdd(&aggr_sca[(size_t)src * 16 + o], nss * ess * coeff);
  float* av = aggr_vec + (size_t)src * 48 + o * 3;
  atomicAdd(&av[0], (nv[0] * esv + nsv * ev0) * coeff);
  atomicAdd(&av[1], (nv[1] * esv + nsv * ev1) * coeff);
  atomicAdd(&av[2], (nv[2] * esv + nsv * ev2) * coeff);
}

// ---------------------------------------------------------------------------
// K3: GVPerceptronVN. One wave32 handles a 16-node tile; all 16x16 matmuls
// run on V_WMMA_F32_16X16X4_F32. D-layout -> A-layout transposes go through
// a per-wave LDS tile (DS ops are in-order within a wave). Aggregates are
// consumed exactly once (NT loads); outputs are written once (NT stores).
__global__ __launch_bounds__(256) void gvp_node_kernel(
    const float* __restrict__ aggr_sca, const float* __restrict__ aggr_vec,
    const float* __restrict__ W_lv, const float* __restrict__ W_lv2,
    const float* __restrict__ W_ls, const float* __restrict__ W_gate,
    const float* __restrict__ b_gate, const float* __restrict__ W_dir,
    float* __restrict__ out_sca, float* __restrict__ out_vec,
    int N, int tiles) {
  __shared__ float lds[WAVES_PER_BLOCK][16][17];
  const int wave = threadIdx.x >> 5;
  const int lane = threadIdx.x & 31;
  const int tile = blockIdx.x * WAVES_PER_BLOCK + wave;
  if (tile >= tiles) return;  // wave-uniform: EXEC stays all-ones for WMMA
  const int base = tile << 4;
  const int l16 = lane & 15;
  const int hi  = lane >> 4;  // 0 or 1
  float (*T)[17] = lds[wave];
  const int nrow = min(base + l16, N - 1);  // clamped A-side node row

  const v8f vzero = {0.f, 0.f, 0.f, 0.f, 0.f, 0.f, 0.f, 0.f};

  // A-fragments of aggr_sca  (A[m][k]: m=l16, k=4q+2*hi+{0,1})
  v2f aS[4];
  {
    const float* p = aggr_sca + (size_t)nrow * 16;
#pragma unroll
    for (int q = 0; q < 4; ++q) {
      const int ch = 4 * q + 2 * hi;
      aS[q].x = ldnt(p + ch);
      aS[q].y = ldnt(p + ch + 1);
    }
  }

  // ---- Phase A: v_inter_i = aggr_vec_i @ W_lv; norm^2; capture A-frags ----
  v2f aVI[3][4];
  v8f nrm2 = vzero;
#pragma unroll
  for (int i = 0; i < 3; ++i) {
    v2f aV[4];
    const float* p = aggr_vec + (size_t)nrow * 48 + i;
#pragma unroll
    for (int q = 0; q < 4; ++q) {
      const int ch = 4 * q + 2 * hi;
      aV[q].x = ldnt(p + ch * 3);
      aV[q].y = ldnt(p + (ch + 1) * 3);
    }
    v8f vi = mm16(aV, W_lv, l16, hi, vzero);
#pragma unroll
    for (int r = 0; r < 8; ++r) nrm2[r] += vi[r] * vi[r];
    // D -> A through LDS (T[node][ch])
#pragma unroll
    for (int r = 0; r < 8; ++r) T[r + 8 * hi][l16] = vi[r];
    __builtin_amdgcn_wave_barrier();
#pragma unroll
    for (int q = 0; q < 4; ++q) {
      const int ch = 4 * q + 2 * hi;
      aVI[i][q].x = T[l16][ch];
      aVI[i][q].y = T[l16][ch + 1];
    }
    __builtin_amdgcn_wave_barrier();
  }

  // ---- Phase B: out_sca = concat(v_norm, aggr_sca) @ W_ls  (K=32) ----
  v8f vn;
#pragma unroll
  for (int r = 0; r < 8; ++r) vn[r] = sqrtf(nrm2[r]);
#pragma unroll
  for (int r = 0; r < 8; ++r) T[r + 8 * hi][l16] = vn[r];
  __builtin_amdgcn_wave_barrier();
  v2f aN[4];
#pragma unroll
  for (int q = 0; q < 4; ++q) {
    const int ch = 4 * q + 2 * hi;
    aN[q].x = T[l16][ch];
    aN[q].y = T[l16][ch + 1];
  }
  __builtin_amdgcn_wave_barrier();
  v8f sD = mm16(aN, W_ls, l16, hi, vzero);        // rows 0..15 (v_norm)
  sD = mm16(aS, W_ls + 256, l16, hi, sD);         // rows 16..31 (aggr_sca)

  // ---- Phase C: gate = sigmoid(out_sca @ W_gate + b_gate) ----
#pragma unroll
  for (int r = 0; r < 8; ++r) T[r + 8 * hi][l16] = sD[r];
  __builtin_amdgcn_wave_barrier();
  v2f aOS[4];
#pragma unroll
  for (int q = 0; q < 4; ++q) {
    const int ch = 4 * q + 2 * hi;
    aOS[q].x = T[l16][ch];
    aOS[q].y = T[l16][ch + 1];
  }
  __builtin_amdgcn_wave_barrier();
  v8f gl = mm16(aOS, W_gate, l16, hi, vzero);
  const float bg = b_gate[l16];
  v8f gate;
#pragma unroll
  for (int r = 0; r < 8; ++r) gate[r] = 1.0f / (1.0f + expf(-(gl[r] + bg)));

  // ---- Phase D: out_vec_i = gate * (v_inter_i @ W_lv2); d_i = ov_i @ W_dir
  v8f ov[3], dv[3], dotv = vzero, dn2 = vzero;
#pragma unroll
  for (int i = 0; i < 3; ++i) {
    v8f t = mm16(aVI[i], W_lv2, l16, hi, vzero);
#pragma unroll
    for (int r = 0; r < 8; ++r) t[r] *= gate[r];
    ov[i] = t;
#pragma unroll
    for (int r = 0; r < 8; ++r) T[r + 8 * hi][l16] = t[r];
    __builtin_amdgcn_wave_barrier();
    v2f aOV[4];
#pragma unroll
    for (int q = 0; q < 4; ++q) {
      const int ch = 4 * q + 2 * hi;
      aOV[q].x = T[l16][ch];
      aOV[q].y = T[l16][ch + 1];
    }
    __builtin_amdgcn_wave_barrier();
    v8f di = mm16(aOV, W_dir, l16, hi, vzero);
    dv[i] = di;
#pragma unroll
    for (int r = 0; r < 8; ++r) {
      dotv[r] += t[r] * di[r];
      dn2[r] += di[r] * di[r];
    }
  }

  // ---- Finalize: VN-LeakyReLU(0.2) on vectors, LeakyReLU(0.01) on scalars
#pragma unroll
  for (int r = 0; r < 8; ++r) {
    const int node = base + r + 8 * hi;  // D-layout row
    if (node < N) {
      const float s = sD[r];
      stnt(out_sca + (size_t)node * 16 + l16, (s >= 0.0f) ? s : 0.01f * s);
      const float dt = dotv[r];
      const float sc = (dt >= 0.0f) ? 0.0f : dt / (dn2[r] + 1e-6f);
      float* op = out_vec + (size_t)node * 48 + l16 * 3;
#pragma unroll
      for (int i = 0; i < 3; ++i)
        stnt(op + i, 0.2f * ov[i][r] + 0.8f * (ov[i][r] - sc * dv[i][r]));
    }
  }
}

// ---------------------------------------------------------------------------
extern "C" void kernel_launch(void* const* d_in, const int* in_sizes, int n_in,
                              void* d_out, int out_size, void* d_ws, size_t ws_size,
                              hipStream_t stream) {
  const float* node_sca  = (const float*)d_in[0];
  const float* node_vec  = (const float*)d_in[1];
  const float* edge_sca  = (const float*)d_in[2];
  const float* edge_vec  = (const float*)d_in[3];
  const float* gds_dist  = (const float*)d_in[4];
  const int*   edge_index= (const int*)  d_in[5];
  const float* W_nss = (const float*)d_in[6];   const float* b_nss = (const float*)d_in[7];
  const float* W_ess = (const float*)d_in[8];   const float* b_ess = (const float*)d_in[9];
  const float* W_nsv = (const float*)d_in[10];  const float* b_nsv = (const float*)d_in[11];
  const float* W_esv = (const float*)d_in[12];  const float* b_esv = (const float*)d_in[13];
  const float* W_nvv = (const float*)d_in[14];
  const float* W_evv = (const float*)d_in[15];
  const float* W_lv  = (const float*)d_in[16];
  const float* W_lv2 = (const float*)d_in[17];
  const float* W_ls  = (const float*)d_in[18];
  const float* W_gate= (const float*)d_in[19];  const float* b_gate= (const float*)d_in[20];
  const float* W_dir = (const float*)d_in[21];

  const int N = in_sizes[0] / 4;  // node_sca is [N,4]
  const int E = in_sizes[4];      // gds_dist is [E]

  float* w = (float*)d_ws;
  float* node_ss  = w;
  float* node_sv  = w + (size_t)N * 16;
  float* node_vv  = w + (size_t)N * 32;
  float* aggr_sca = w + (size_t)N * 80;
  float* aggr_vec = w + (size_t)N * 96;  // aggr region is N*64 floats contiguous

  float* out_sca = (float*)d_out;
  float* out_vec = out_sca + (size_t)N * 16;

  zero_kernel<<<2048, 256, 0, stream>>>(aggr_sca, (long)N * 64);

  node_proj_kernel<<<(N * 16 + 255) / 256, 256, 0, stream>>>(
      node_sca, node_vec, W_nss, b_nss, W_nsv, b_nsv, W_nvv,
      node_ss, node_sv, node_vv, N);

  edge_msg_kernel<<<(E * 16 + 255) / 256, 256, 0, stream>>>(
      edge_sca, edge_vec, gds_dist, edge_index,
      W_ess, b_ess, W_esv, b_esv, W_evv,
      node_ss, node_sv, node_vv, aggr_sca, aggr_vec, E);

  const int tiles = (N + 15) / 16;
  gvp_node_kernel<<<(tiles + WAVES_PER_BLOCK - 1) / WAVES_PER_BLOCK, 256, 0, stream>>>(
      aggr_sca, aggr_vec, W_lv, W_lv2, W_ls, W_gate, b_gate, W_dir,
      out_sca, out_vec, N, tiles);
}